// CenterGAT_QNet_55576876810745
// MI455X (gfx1250) — compile-verified
//
#include <hip/hip_runtime.h>
#include <cstdint>

#define BB 8
#define NN 256
#define HH 64
#define AA 10
#define LL 2
#define EPSV 1e-5f

typedef __attribute__((ext_vector_type(2))) float v2f;
typedef __attribute__((ext_vector_type(8))) float v8f;
typedef int v4i __attribute__((vector_size(16)));  // async-LDS builtin pointee type

// CDNA5 async global->LDS path (ASYNCcnt-tracked), if the toolchain exposes it.
#if defined(__has_builtin)
#if __has_builtin(__builtin_amdgcn_global_load_async_to_lds_b128) && \
    __has_builtin(__builtin_amdgcn_s_wait_asynccnt)
#define USE_ASYNC_LDS 1
#endif
#endif

// XOR swizzle on 4-float groups: kills 16-way LDS bank conflicts on the
// column-strided A-fragment reads while keeping float4 groups contiguous.
__device__ __forceinline__ int eidx(int m, int k) {
  return m * HH + (k ^ ((m & 15) << 2));
}

// ---------------------------------------------------------------------------
// Per-layer column sums: w3sum[k] = sum_h W3[h,k]; w4sum[k] = sum_h W4[h,k]
// ---------------------------------------------------------------------------
__global__ void colsum_kernel(const float* __restrict__ W3,
                              const float* __restrict__ W4,
                              float* __restrict__ w3sum,
                              float* __restrict__ w4sum) {
  const int k = threadIdx.x;
  float a = 0.f, c = 0.f;
  for (int h = 0; h < HH; ++h) {
    a += W3[h * HH + k];
    c += W4[h * HH + k];
  }
  w3sum[k] = a;
  w4sum[k] = c;
}

// ---------------------------------------------------------------------------
// Node projections: Xj = X@W5^T, Xres = X@W6^T, s = X . w3sum   (tiny)
// ---------------------------------------------------------------------------
__global__ __launch_bounds__(256) void node_proj_kernel(
    const float* __restrict__ X, const float* __restrict__ W5,
    const float* __restrict__ W6, const float* __restrict__ w3sum,
    float* __restrict__ s, float* __restrict__ Xj, float* __restrict__ Xres) {
  __shared__ float sX[256];  // 4 node rows per block
  const int tid = threadIdx.x;
  const int t = blockIdx.x * 256 + tid;  // linear over (b,i,h)
  sX[tid] = X[t];
  __syncthreads();
  const int h = t & 63;
  const float* x = sX + ((tid >> 6) << 6);
  float aj = 0.f, ar = 0.f;
  #pragma unroll 8
  for (int k = 0; k < HH; ++k) {
    aj += x[k] * W5[h * HH + k];
    ar += x[k] * W6[h * HH + k];
  }
  Xj[t] = aj;
  Xres[t] = ar;
  if (h == 0) {
    float sv = 0.f;
    for (int k = 0; k < HH; ++k) sv += x[k] * w3sum[k];
    s[t >> 6] = sv;
  }
}

// ---------------------------------------------------------------------------
// Fused edge layer: one workgroup per (b,i).
//   async E-tile stage -> logits -> softmax -> E_hat = E_tile @ W4^T (WMMA
//   fp32) -> att scale -> agg reduction -> edge LN store -> fused X update.
// ---------------------------------------------------------------------------
__global__ __launch_bounds__(256) void layer_edge_kernel(
    const float* __restrict__ Xin, const float* __restrict__ Ein,
    const float* __restrict__ W4,
    const float* __restrict__ svec, const float* __restrict__ w4sum,
    const float* __restrict__ Xj, const float* __restrict__ Xres,
    const float* __restrict__ ln_ng, const float* __restrict__ ln_nb,
    const float* __restrict__ ln_eg, const float* __restrict__ ln_eb,
    float* __restrict__ Eout, float* __restrict__ Xout, int storeE) {
  __shared__ float sE[NN * HH];   // 64 KB edge tile (swizzled), reused for E_next
  __shared__ float sW[HH * HH];   // 16 KB, sW[k*64+n] = W4[n,k]  (B-fragment layout)
  __shared__ float sAtt[NN];
  __shared__ float sRed[NN];
  __shared__ float sAgg[HH];
  __shared__ float sV[HH];
  __shared__ float sW4s[HH];
  __shared__ float sLN[4 * HH];   // eg | eb | ng | nb

  const int tid = threadIdx.x;
  const int b = blockIdx.x >> 8;
  const int i = blockIdx.x & 255;
  const float* Etile = Ein + ((size_t)(b * NN + i)) * (NN * HH);

  // ---- stage E tile into LDS ----
#ifdef USE_ASYNC_LDS
  // Async DMA: global -> LDS without touching VGPRs; overlaps with the
  // W4 transpose staging below. One B128 per lane per step; the XOR swizzle
  // keeps every 4-float group contiguous so B128 granularity is exact.
  #pragma unroll
  for (int c = 0; c < 16; ++c) {
    int idx = c * 256 + tid;          // float4 index within tile
    int m = idx >> 4;
    int kg = (idx & 15) << 2;
    __builtin_amdgcn_global_load_async_to_lds_b128(
        (v4i*)(Etile + (size_t)idx * 4),
        (v4i*)(&sE[eidx(m, kg)]), 0, 0);
  }
#else
  for (int c = 0; c < 16; ++c) {
    int idx = c * 256 + tid;          // float4 index within tile
    int m = idx >> 4;
    int kg = (idx & 15) << 2;
    float4 v = reinterpret_cast<const float4*>(Etile)[idx];
    int base = eidx(m, kg);
    sE[base + 0] = v.x; sE[base + 1] = v.y;
    sE[base + 2] = v.z; sE[base + 3] = v.w;
  }
#endif

  // ---- stage W4^T + small vectors (overlaps with async E fill) ----
  for (int c = 0; c < 16; ++c) {
    int idx = c * 256 + tid;          // W4 row-major element h*64+k
    int h = idx >> 6, k = idx & 63;
    sW[k * HH + h] = W4[idx];
  }
  if (tid < HH) {
    sW4s[tid] = w4sum[tid];
    sAgg[tid] = 0.f;
    sLN[tid]          = ln_eg[tid];
    sLN[HH + tid]     = ln_eb[tid];
    sLN[2 * HH + tid] = ln_ng[tid];
    sLN[3 * HH + tid] = ln_nb[tid];
  }
#ifdef USE_ASYNC_LDS
  __builtin_amdgcn_s_wait_asynccnt(0);
#endif
  __syncthreads();

  // ---- attention logits + softmax over j (thread j = tid) ----
  float logit;
  {
    float acc = 0.f;
    #pragma unroll 8
    for (int k = 0; k < HH; ++k) acc += sE[eidx(tid, k)] * sW4s[k];
    logit = svec[b * NN + i] + svec[b * NN + tid] + acc;
  }
  sRed[tid] = logit;
  __syncthreads();
  for (int off = 128; off > 0; off >>= 1) {
    if (tid < off) sRed[tid] = fmaxf(sRed[tid], sRed[tid + off]);
    __syncthreads();
  }
  const float mx = sRed[0];
  __syncthreads();
  const float ex = __expf(logit - mx);
  sRed[tid] = ex;
  __syncthreads();
  for (int off = 128; off > 0; off >>= 1) {
    if (tid < off) sRed[tid] += sRed[tid + off];
    __syncthreads();
  }
  const float inv = 1.f / sRed[0];
  sAtt[tid] = ex * inv;
  __syncthreads();

  // ---- WMMA GEMM: E_hat(256x64) = E_tile(256x64) @ W4^T(64x64) ----
  const int wave = tid >> 5;
  const int lane = tid & 31;
  const int lm = lane & 15;
  const int lh = lane >> 4;  // K-half select per A/B layout
  const float* XjB = Xj + b * (NN * HH);

  for (int slab = 0; slab < 2; ++slab) {
    const int m0 = (wave * 2 + slab) * 16;
    v8f c0 = {}, c1 = {}, c2 = {}, c3 = {};
    #pragma unroll
    for (int kb = 0; kb < 16; ++kb) {
      const int k = kb * 4 + lh * 2;
      v2f a;  a.x  = sE[eidx(m0 + lm, k)];   a.y  = sE[eidx(m0 + lm, k + 1)];
      v2f b0; b0.x = sW[k * HH + lm];        b0.y = sW[(k + 1) * HH + lm];
      v2f b1; b1.x = sW[k * HH + 16 + lm];   b1.y = sW[(k + 1) * HH + 16 + lm];
      v2f b2; b2.x = sW[k * HH + 32 + lm];   b2.y = sW[(k + 1) * HH + 32 + lm];
      v2f b3; b3.x = sW[k * HH + 48 + lm];   b3.y = sW[(k + 1) * HH + 48 + lm];
      c0 = __builtin_amdgcn_wmma_f32_16x16x4_f32(false, a, false, b0, (short)0, c0, false, false);
      c1 = __builtin_amdgcn_wmma_f32_16x16x4_f32(false, a, false, b1, (short)0, c1, false, false);
      c2 = __builtin_amdgcn_wmma_f32_16x16x4_f32(false, a, false, b2, (short)0, c2, false, false);
      c3 = __builtin_amdgcn_wmma_f32_16x16x4_f32(false, a, false, b3, (short)0, c3, false, false);
    }
    // att scale, agg partials, write E_next back into own rows of sE
    float p0 = 0.f, p1 = 0.f, p2 = 0.f, p3 = 0.f;
    #pragma unroll
    for (int r = 0; r < 8; ++r) {
      const int j = m0 + r + 8 * lh;  // D layout: M = r + 8*(lane>=16)
      const float at = sAtt[j];
      const float v0 = c0[r] * at, v1 = c1[r] * at;
      const float v2 = c2[r] * at, v3 = c3[r] * at;
      const float* xjr = XjB + j * HH;
      p0 += v0 * xjr[lm];      p1 += v1 * xjr[16 + lm];
      p2 += v2 * xjr[32 + lm]; p3 += v3 * xjr[48 + lm];
      sE[eidx(j, lm)] = v0;      sE[eidx(j, 16 + lm)] = v1;
      sE[eidx(j, 32 + lm)] = v2; sE[eidx(j, 48 + lm)] = v3;
    }
    atomicAdd(&sAgg[lm], p0);      atomicAdd(&sAgg[16 + lm], p1);
    atomicAdd(&sAgg[32 + lm], p2); atomicAdd(&sAgg[48 + lm], p3);
  }
  __syncthreads();

  // ---- edge layernorm + store (skipped for last layer: E is dead) ----
  if (storeE) {
    const int j = tid;
    float mean = 0.f;
    #pragma unroll 8
    for (int k = 0; k < HH; ++k) mean += sE[eidx(j, k)];
    mean *= (1.f / HH);
    float var = 0.f;
    #pragma unroll 8
    for (int k = 0; k < HH; ++k) { float d = sE[eidx(j, k)] - mean; var += d * d; }
    var *= (1.f / HH);
    const float rstd = rsqrtf(var + EPSV);
    float* orow = Eout + (((size_t)(b * NN + i)) * NN + j) * HH;
    #pragma unroll
    for (int kg = 0; kg < 16; ++kg) {
      const int k = kg * 4;
      float4 o;
      o.x = (sE[eidx(j, k + 0)] - mean) * rstd * sLN[k + 0] + sLN[HH + k + 0];
      o.y = (sE[eidx(j, k + 1)] - mean) * rstd * sLN[k + 1] + sLN[HH + k + 1];
      o.z = (sE[eidx(j, k + 2)] - mean) * rstd * sLN[k + 2] + sLN[HH + k + 2];
      o.w = (sE[eidx(j, k + 3)] - mean) * rstd * sLN[k + 3] + sLN[HH + k + 3];
      reinterpret_cast<float4*>(orow)[kg] = o;
    }
  }

  // ---- fused X update: LN(relu(agg + Xres)) + X ----
  if (tid < HH) {
    const float v = fmaxf(sAgg[tid] + Xres[(b * NN + i) * HH + tid], 0.f);
    sV[tid] = v;
  }
  __syncthreads();
  if (tid < HH) {
    float mean = 0.f;
    for (int k = 0; k < HH; ++k) mean += sV[k];
    mean *= (1.f / HH);
    float var = 0.f;
    for (int k = 0; k < HH; ++k) { float d = sV[k] - mean; var += d * d; }
    var *= (1.f / HH);
    const float rstd = rsqrtf(var + EPSV);
    const int off = (b * NN + i) * HH + tid;
    Xout[off] = (sV[tid] - mean) * rstd * sLN[2 * HH + tid] + sLN[3 * HH + tid] + Xin[off];
  }
}

// ---------------------------------------------------------------------------
// Head MLP: out = relu(X@hw1^T + hb1) @ hw2^T + hb2
// ---------------------------------------------------------------------------
__global__ __launch_bounds__(256) void head_kernel(
    const float* __restrict__ X, const float* __restrict__ hw1,
    const float* __restrict__ hb1, const float* __restrict__ hw2,
    const float* __restrict__ hb2, float* __restrict__ out) {
  __shared__ float s1[HH * HH];
  __shared__ float s2[AA * HH];
  __shared__ float sb1[HH];
  __shared__ float sb2[AA];
  const int tid = threadIdx.x;
  for (int c = 0; c < 16; ++c) s1[c * 256 + tid] = hw1[c * 256 + tid];
  for (int idx = tid; idx < AA * HH; idx += 256) s2[idx] = hw2[idx];
  if (tid < HH) sb1[tid] = hb1[tid];
  if (tid < AA) sb2[tid] = hb2[tid];
  __syncthreads();
  const int bi = blockIdx.x * 256 + tid;
  const float* xr = X + bi * HH;
  float x[HH];
  #pragma unroll
  for (int k = 0; k < HH; ++k) x[k] = xr[k];
  float o[AA];
  #pragma unroll
  for (int a = 0; a < AA; ++a) o[a] = sb2[a];
  for (int u = 0; u < HH; ++u) {
    float hu = sb1[u];
    #pragma unroll 8
    for (int k = 0; k < HH; ++k) hu += x[k] * s1[u * HH + k];
    hu = fmaxf(hu, 0.f);
    #pragma unroll
    for (int a = 0; a < AA; ++a) o[a] += hu * s2[a * HH + u];
  }
  for (int a = 0; a < AA; ++a) out[bi * AA + a] = o[a];
}

// ---------------------------------------------------------------------------
extern "C" void kernel_launch(void* const* d_in, const int* in_sizes, int n_in,
                              void* d_out, int out_size, void* d_ws, size_t ws_size,
                              hipStream_t stream) {
  (void)in_sizes; (void)n_in; (void)out_size; (void)ws_size;
  const float* X     = (const float*)d_in[0];
  const float* E     = (const float*)d_in[1];
  const float* W3    = (const float*)d_in[2];
  const float* W4    = (const float*)d_in[3];
  const float* W5    = (const float*)d_in[4];
  const float* W6    = (const float*)d_in[5];
  const float* ln_ng = (const float*)d_in[6];
  const float* ln_nb = (const float*)d_in[7];
  const float* ln_eg = (const float*)d_in[8];
  const float* ln_eb = (const float*)d_in[9];
  const float* hw1   = (const float*)d_in[10];
  const float* hb1   = (const float*)d_in[11];
  const float* hw2   = (const float*)d_in[12];
  const float* hb2   = (const float*)d_in[13];

  float* ws = (float*)d_ws;
  const size_t nE = (size_t)BB * NN * NN * HH;  // 33.5M floats (134 MB)
  const size_t nX = (size_t)BB * NN * HH;       // 131072 floats
  float* Ebuf  = ws; ws += nE;
  float* Xbuf0 = ws; ws += nX;
  float* Xbuf1 = ws; ws += nX;
  float* XjB   = ws; ws += nX;
  float* XresB = ws; ws += nX;
  float* sB    = ws; ws += (size_t)BB * NN;
  float* w3s   = ws; ws += HH;
  float* w4s   = ws; ws += HH;

  const float* Xcur = X;
  const float* Ecur = E;
  float* Xnexts[2] = {Xbuf0, Xbuf1};
  for (int l = 0; l < LL; ++l) {
    colsum_kernel<<<1, HH, 0, stream>>>(W3 + l * HH * HH, W4 + l * HH * HH, w3s, w4s);
    node_proj_kernel<<<(BB * NN * HH) / 256, 256, 0, stream>>>(
        Xcur, W5 + l * HH * HH, W6 + l * HH * HH, w3s, sB, XjB, XresB);
    layer_edge_kernel<<<BB * NN, 256, 0, stream>>>(
        Xcur, Ecur, W4 + l * HH * HH, sB, w4s, XjB, XresB,
        ln_ng + l * HH, ln_nb + l * HH, ln_eg + l * HH, ln_eb + l * HH,
        Ebuf, Xnexts[l], (l == 0) ? 1 : 0);
    Xcur = Xnexts[l];
    Ecur = Ebuf;
  }
  head_kernel<<<BB, 256, 0, stream>>>(Xcur, hw1, hb1, hw2, hb2, (float*)d_out);
}